// ROIAlign_89129161326780
// MI455X (gfx1250) — compile-verified
//
#include <hip/hip_runtime.h>
#include <hip/hip_bf16.h>

// ROIAlign forward for MI455X (gfx1250).
// Strategy: per (roi, 8-channel chunk) block, stage the ROI's pixel window as a
// 3D tile (cols x rows x channels) into LDS with one Tensor Data Mover
// tensor_load_to_lds, then do all bilinear gathers from LDS.
// wave32 layout: lane = {sxi = lane&15 (14 x-samples), half = lane>>4 (y corner)}.

#define OUT_HW 7
#define SR 2
#define NS 14            // OUT_HW * SR samples per axis
#define CPB 8            // channels per block (TDM tile z-dim)
#define KMAX 44          // max window extent (max roi span 40 + corners + slack)

typedef __attribute__((ext_vector_type(4))) unsigned int v4u;
typedef __attribute__((ext_vector_type(8))) int          v8i;
typedef __attribute__((ext_vector_type(4))) int          v4i;

struct SmemT {
  float tile[CPB * KMAX * KMAX];   // TDM destination, packed [c][row][col]
  int   xcol[16];                  // window-relative col0 per x-sample (padded to 16)
  float xwa[16], xwb[16];          // x corner weights (validity/edge folded in)
  int   yr0[NS], yr1[NS];          // window-relative rows per y-sample
  float ywa[NS], ywb[NS];          // 0.25 * y weights (validity folded in)
  int   kw, kh;                    // dynamic window dims
};

__global__ __launch_bounds__(256)
void ROIAlign_tdm_kernel(const float* __restrict__ in,
                         const float* __restrict__ rois,
                         float* __restrict__ out,
                         int C, int H, int W)
{
  __shared__ SmemT s;
  const int r     = blockIdx.x;
  const int cBase = blockIdx.y * CPB;
  const int tid   = threadIdx.x;

  if (tid == 0) {
    const float scale = 0.0625f;
    int   b  = (int)rois[r * 5 + 0];
    float x1 = rois[r * 5 + 1] * scale;
    float y1 = rois[r * 5 + 2] * scale;
    float x2 = rois[r * 5 + 3] * scale;
    float y2 = rois[r * 5 + 4] * scale;
    float bw = fmaxf(x2 - x1, 1.0f) * (1.0f / OUT_HW);
    float bh = fmaxf(y2 - y1, 1.0f) * (1.0f / OUT_HW);

    int cmin = W, cmax = 0, rmin = H, rmax = 0;
    for (int k = 0; k < NS; ++k) {
      int p = k >> 1, i = k & 1;
      // ---- X axis sample k ----
      {
        float cx   = x1 + (float)p * bw + ((float)i + 0.5f) * bw * 0.5f;
        bool  vx   = (cx >= -1.0f) && (cx <= (float)W);
        float cc   = fmaxf(cx, 0.0f);
        float lof  = floorf(cc);
        bool  edge = lof >= (float)(W - 1);
        int   lo   = min((int)lof, W - 1);
        float f    = edge ? 0.0f : (cc - lof);
        int   c0   = min(lo, W - 2);               // safe pair load at (c0, c0+1)
        float wa   = edge ? 0.0f : (1.0f - f);     // edge: value lives at c0+1
        float wb   = edge ? 1.0f : f;
        if (!vx) { wa = 0.0f; wb = 0.0f; }
        s.xcol[k] = c0; s.xwa[k] = wa; s.xwb[k] = wb;
        cmin = min(cmin, c0); cmax = max(cmax, c0 + 1);
      }
      // ---- Y axis sample k ----
      {
        float cy   = y1 + (float)p * bh + ((float)i + 0.5f) * bh * 0.5f;
        bool  vy   = (cy >= -1.0f) && (cy <= (float)H);
        float cc   = fmaxf(cy, 0.0f);
        float lof  = floorf(cc);
        bool  edge = lof >= (float)(H - 1);
        int   lo   = min((int)lof, H - 1);
        float f    = edge ? 0.0f : (cc - lof);
        int   r0   = lo, r1 = min(lo + 1, H - 1);  // at edge f==0 -> r1 weight 0
        s.yr0[k] = r0; s.yr1[k] = r1;
        s.ywa[k] = vy ? 0.25f * (1.0f - f) : 0.0f; // fold /(S*S)=0.25 + validity
        s.ywb[k] = vy ? 0.25f * f          : 0.0f;
        rmin = min(rmin, r0); rmax = max(rmax, r1);
      }
    }
    int kw = cmax - cmin + 1;  if (kw > KMAX) kw = KMAX;
    int kh = rmax - rmin + 1;  if (kh > KMAX) kh = KMAX;
    for (int k = 0; k < NS; ++k) {            // make window-relative (defensive clamps)
      s.xcol[k] = min(s.xcol[k] - cmin, kw - 2);
      s.yr0[k]  = min(s.yr0[k]  - rmin, kh - 1);
      s.yr1[k]  = min(s.yr1[k]  - rmin, kh - 1);
    }
    s.xcol[14] = 0; s.xwa[14] = 0.0f; s.xwb[14] = 0.0f;   // pad lanes 14/15
    s.xcol[15] = 0; s.xwa[15] = 0.0f; s.xwb[15] = 0.0f;
    s.kw = kw; s.kh = kh;

    // ---- Build Tensor DMA descriptor (D#) and issue TDM load ----
    unsigned long long gaddr = (unsigned long long)(uintptr_t)in
        + 4ull * (((unsigned long long)(((long long)b * C + cBase) * H + rmin)) * (unsigned)W
                  + (unsigned)cmin);
    unsigned int ldsOff = (unsigned int)(uintptr_t)(void*)&s.tile[0]; // flat->LDS offset (addr[31:0])

    v4u g0;                                   // D# group 0 (128b)
    g0.x = 1u;                                // count=1, is_restore=0, gather_mode=0
    g0.y = ldsOff;                            // lds_addr (bytes)
    g0.z = (unsigned int)gaddr;               // global_addr[31:0]
    g0.w = (unsigned int)(gaddr >> 32) | (2u << 30); // global_addr[56:32] | type=2

    const unsigned int TD = 1u << 24;         // tensor dims: large -> no OOB zeroing
    unsigned int str1 = (unsigned int)(H * W);
    v8i g1;                                   // D# group 1 (256b)
    g1[0] = (int)(2u << 16);                  // wg_mask=0, data_size=2 (4B), flags=0
    g1[1] = (int)((TD & 0xFFFFu) << 16);                              // tensor_dim0[15:0]
    g1[2] = (int)(((TD >> 16) & 0xFFFFu) | ((TD & 0xFFFFu) << 16));   // td0 hi | td1 lo
    g1[3] = (int)(((TD >> 16) & 0xFFFFu) | ((unsigned)s.kw << 16));   // td1 hi | tile_dim0
    g1[4] = (int)((unsigned)s.kh | ((unsigned)CPB << 16));            // tile_dim1 | tile_dim2
    g1[5] = (int)(unsigned)W;                                         // dim0_stride[31:0] = W
    g1[6] = (int)((str1 & 0xFFFFu) << 16);                            // d0str hi16=0 | dim1_stride[15:0]
    g1[7] = (int)(str1 >> 16);                                        // dim1_stride[47:16] (=H*W>>16)
    v4i g2;                                   // D# group 2 (128b), normal mode
    g2[0] = (int)TD;                          // tensor_dim2 (channels; large)
    g2[1] = 0; g2[2] = 0; g2[3] = 0;          // dim3/stride2/tile_dim3 unused
    v4i g3; g3[0] = 0; g3[1] = 0; g3[2] = 0; g3[3] = 0;
    v8i gz;                                   // extra group (clang-23 6-arg form): zeroed
    gz[0] = 0; gz[1] = 0; gz[2] = 0; gz[3] = 0;
    gz[4] = 0; gz[5] = 0; gz[6] = 0; gz[7] = 0;

    __builtin_amdgcn_tensor_load_to_lds(g0, g1, g2, g3, gz, 0);
    __builtin_amdgcn_s_wait_tensorcnt(0);
    __asm__ volatile("" ::: "memory");
  }
  __syncthreads();

  // ---- Compute from LDS tile ----
  const int wav  = tid >> 5;        // wave id -> local channel
  const int lane = tid & 31;
  const int sxi  = lane & 15;
  const int half = lane >> 4;       // 0: y-low corner, 1: y-high corner
  const int c    = cBase + wav;
  const int kw   = s.kw, kh = s.kh;

  const float wxa = s.xwa[sxi];
  const float wxb = s.xwb[sxi];
  const int   xc  = s.xcol[sxi];
  const int   tb  = wav * (kh * kw);

  int   off[NS];
  float wy[NS];
#pragma unroll
  for (int k = 0; k < NS; ++k) {
    int row = half ? s.yr1[k] : s.yr0[k];
    wy[k]   = half ? s.ywb[k] : s.ywa[k];
    off[k]  = tb + row * kw + xc;
  }

  const long long obase = (((long long)r * C) + c) * (OUT_HW * OUT_HW);
#pragma unroll
  for (int ph = 0; ph < OUT_HW; ++ph) {
    float acc = 0.0f;
#pragma unroll
    for (int sy = 0; sy < SR; ++sy) {
      const int k = ph * SR + sy;
      float a = s.tile[off[k]];
      float b = s.tile[off[k] + 1];
      acc = fmaf(wy[k], fmaf(wxa, a, wxb * b), acc);
    }
    acc += __shfl_xor(acc, 16, 32);   // combine y corners (wave32 halves)
    acc += __shfl_xor(acc, 1, 32);    // combine the two x samples of a bin
    if (half == 0 && sxi < 14 && (sxi & 1) == 0) {
      out[obase + ph * OUT_HW + (sxi >> 1)] = acc;  // 7 lanes, contiguous 28B
    }
  }
}

extern "C" void kernel_launch(void* const* d_in, const int* in_sizes, int n_in,
                              void* d_out, int out_size, void* d_ws, size_t ws_size,
                              hipStream_t stream) {
  const float* in   = (const float*)d_in[0];
  const float* rois = (const float*)d_in[1];
  float*       o    = (float*)d_out;
  const int C = 256, H = 200, W = 200;
  const int R = in_sizes[1] / 5;          // 512
  dim3 grid((unsigned)R, (unsigned)(C / CPB));   // 512 x 32 blocks
  ROIAlign_tdm_kernel<<<grid, 256, 0, stream>>>(in, rois, o, C, H, W);
}